// CALayer_661424963697
// MI455X (gfx1250) — compile-verified
//
#include <hip/hip_runtime.h>
#include <hip/hip_bf16.h>

typedef __attribute__((ext_vector_type(2))) float v2f;
typedef __attribute__((ext_vector_type(8))) float v8f;

#define C_DIM 256
#define CR_DIM 16
#define HW 16384          // 128*128
#define HW4 4096          // HW/4

// ---------------------------------------------------------------------------
// Kernel 1: global average pool. One block per (b,c) plane (4096 blocks).
// 16384 floats per plane, float4 coalesced, LDS tree reduce.
// ---------------------------------------------------------------------------
__global__ void ca_pool_kernel(const float* __restrict__ x,
                               float* __restrict__ pooled) {
    const float4* xp = (const float4*)(x + (size_t)blockIdx.x * HW);
    float s = 0.0f;
    #pragma unroll 4
    for (int j = threadIdx.x; j < HW4; j += 256) {
        float4 v = xp[j];
        s += (v.x + v.y) + (v.z + v.w);
    }
    __shared__ float red[256];
    red[threadIdx.x] = s;
    __syncthreads();
    #pragma unroll
    for (int off = 128; off > 0; off >>= 1) {
        if (threadIdx.x < off) red[threadIdx.x] += red[threadIdx.x + off];
        __syncthreads();
    }
    if (threadIdx.x == 0) pooled[blockIdx.x] = red[0] * (1.0f / (float)HW);
}

// ---------------------------------------------------------------------------
// Kernel 2: the SE MLP on one wave32 using V_WMMA_F32_16X16X4_F32.
//   mm1: H = relu(P[16,256] x W1^T[256,16] + b1)   -> 64 WMMAs (K=256)
//   mm2: G = sigmoid(H[16,16] x W2^T[16,256] + b2) -> 16 N-tiles * 4 = 64 WMMAs
// A layout (16x4 f32): lanes 0-15 hold M=lane, VGPR0/1 = K,K+1; lanes 16-31
// hold same M with K+2,K+3. B layout (4x16): lane = N, VGPRs stripe K the
// same way. C/D: 8 VGPRs, lane = N (mod 16), VGPR j = row M=j (+8 for lanes>=16).
// ---------------------------------------------------------------------------
__global__ void ca_mlp_kernel(const float* __restrict__ pooled,  // [16,256]
                              const float* __restrict__ w1,      // [16,256]
                              const float* __restrict__ b1,      // [16]
                              const float* __restrict__ w2,      // [256,16]
                              const float* __restrict__ b2,      // [256]
                              float* __restrict__ gates) {       // [16,256]
    const int lane = threadIdx.x;     // 0..31
    const int half = lane >> 4;       // 0/1 -> K offset 0/2
    const int l16  = lane & 15;       // M for A-frags, N for B/C/D-frags

    __shared__ float hlds[16 * 16];

    // ---- mm1: accumulate over K=256 in steps of 4 ----
    v8f acc = {};
    for (int kk = 0; kk < C_DIM; kk += 4) {
        const int k = kk + 2 * half;
        v2f a, b;
        a.x = pooled[l16 * C_DIM + k];      // A[M=l16][k]
        a.y = pooled[l16 * C_DIM + k + 1];
        b.x = w1[l16 * C_DIM + k];          // B[k][N=l16] = w1[N][k]
        b.y = w1[l16 * C_DIM + k + 1];
        acc = __builtin_amdgcn_wmma_f32_16x16x4_f32(
            false, a, false, b, (short)0, acc, false, false);
    }

    // bias + relu, stage H into LDS as H[M][N] row-major
    {
        const float bias = b1[l16];         // bias indexed by N
        #pragma unroll
        for (int j = 0; j < 8; ++j) {
            float v = acc[j] + bias;
            v = v > 0.0f ? v : 0.0f;
            const int m = j + 8 * half;
            hlds[m * 16 + l16] = v;
        }
    }
    __syncthreads();

    // Preload the 4 A-fragments of H (16x16, K-chunks of 4)
    v2f afrag[4];
    #pragma unroll
    for (int t = 0; t < 4; ++t) {
        const int k = 4 * t + 2 * half;
        afrag[t].x = hlds[l16 * 16 + k];
        afrag[t].y = hlds[l16 * 16 + k + 1];
    }

    // ---- mm2: 16 N-tiles of 16 output channels each ----
    for (int nt = 0; nt < 16; ++nt) {
        const int n = nt * 16 + l16;        // output channel for this lane
        v8f acc2 = {};
        #pragma unroll
        for (int t = 0; t < 4; ++t) {
            const int k = 4 * t + 2 * half;
            v2f b;
            b.x = w2[n * CR_DIM + k];       // B[k][n] = w2[n][k]
            b.y = w2[n * CR_DIM + k + 1];
            acc2 = __builtin_amdgcn_wmma_f32_16x16x4_f32(
                false, afrag[t], false, b, (short)0, acc2, false, false);
        }
        const float bias2 = b2[n];
        #pragma unroll
        for (int j = 0; j < 8; ++j) {
            const float v = acc2[j] + bias2;
            const float g = 1.0f / (1.0f + __expf(-v));   // sigmoid
            const int bb = j + 8 * half;                  // batch row
            gates[bb * C_DIM + n] = g;
        }
    }
}

// ---------------------------------------------------------------------------
// Kernel 3: out = x * gate[b,c], float4 streaming. channel = flat >> 14.
// ---------------------------------------------------------------------------
__global__ void ca_gate_kernel(const float* __restrict__ x,
                               const float* __restrict__ gates,
                               float* __restrict__ out) {
    const size_t i4 = (size_t)blockIdx.x * blockDim.x + threadIdx.x;
    const size_t base = i4 * 4;
    const int ch = (int)(base >> 14);       // b*256 + c
    const float g = gates[ch];
    float4 v = ((const float4*)x)[i4];
    v.x *= g; v.y *= g; v.z *= g; v.w *= g;
    ((float4*)out)[i4] = v;
}

// ---------------------------------------------------------------------------
extern "C" void kernel_launch(void* const* d_in, const int* in_sizes, int n_in,
                              void* d_out, int out_size, void* d_ws, size_t ws_size,
                              hipStream_t stream) {
    const float* x  = (const float*)d_in[0];   // [16,256,128,128]
    const float* w1 = (const float*)d_in[1];   // [16,256]
    const float* b1 = (const float*)d_in[2];   // [16]
    const float* w2 = (const float*)d_in[3];   // [256,16]
    const float* b2 = (const float*)d_in[4];   // [256]
    float* out = (float*)d_out;

    float* pooled = (float*)d_ws;              // 4096 floats
    float* gates  = pooled + 4096;             // 4096 floats

    ca_pool_kernel<<<4096, 256, 0, stream>>>(x, pooled);
    ca_mlp_kernel<<<1, 32, 0, stream>>>(pooled, w1, b1, w2, b2, gates);

    const int total4 = (16 * C_DIM * HW) / 4;  // 16,777,216 float4
    ca_gate_kernel<<<total4 / 256, 256, 0, stream>>>(x, gates, out);
}